// Crosscorrelation_48180943126728
// MI455X (gfx1250) — compile-verified
//
#include <hip/hip_runtime.h>

typedef float v2f __attribute__((ext_vector_type(2)));
typedef float v4f __attribute__((ext_vector_type(4)));
typedef float v8f __attribute__((ext_vector_type(8)));

#define WAVE 32

// Streams x,y and accumulates sum(x*y) partials per block into `partial`.
// The MAC engine is V_WMMA_F32_16X16X4_F32: with identical index maps on the
// A and B operands, the diagonal of the 16x16 f32 accumulator holds exact
// per-column dot-product partials; off-diagonal entries are never read.
__global__ __launch_bounds__(256) void cc_dot_wmma_kernel(
    const float* __restrict__ x, const float* __restrict__ y,
    float* __restrict__ partial, long long n)
{
    const int  lane        = threadIdx.x & (WAVE - 1);
    const int  waveInBlock = (int)(threadIdx.x >> 5);
    const int  wavesPerBlk = (int)(blockDim.x >> 5);
    const long long waveGlobal = (long long)blockIdx.x * wavesPerBlk + waveInBlock;
    const long long totalWaves = (long long)gridDim.x * wavesPerBlk;

    const int m  = lane & 15;
    const int hi = lane >> 4;
    const long long laneOff = (long long)m * 8 + (long long)hi * 4;

    const long long nChunks = n >> 7;   // 128 elements per wave-chunk

    v8f acc = {};
    for (long long c = waveGlobal; c < nChunks; c += totalWaves) {
        const long long base = (c << 7) + laneOff;
        v4f xa = *(const v4f*)(x + base);   // global_load_b128, wave covers 512B
        v4f ya = *(const v4f*)(y + base);
        v2f a0 = { xa.x, xa.y }, b0 = { ya.x, ya.y };
        v2f a1 = { xa.z, xa.w }, b1 = { ya.z, ya.w };
        // 8 args: (neg_a, A, neg_b, B, c_mod, C, reuse_a, reuse_b)
        acc = __builtin_amdgcn_wmma_f32_16x16x4_f32(false, a0, false, b0,
                                                    (short)0, acc, false, false);
        acc = __builtin_amdgcn_wmma_f32_16x16x4_f32(false, a1, false, b1,
                                                    (short)0, acc, false, false);
    }

    // Diagonal extraction per C/D layout:
    //   VGPR v: lanes 0-15 -> (M=v, N=lane); lanes 16-31 -> (M=v+8, N=lane-16)
    //   (n,n) for n<8  lives in lane n,    acc[n]
    //   (n,n) for n>=8 lives in lane n+16, acc[n-8]
    float d = 0.f;
#pragma unroll
    for (int v = 0; v < 8; ++v) {
        float cv = acc[v];
        d = (lane == v || lane == v + 24) ? cv : d;
    }

    // Scalar tail for n not divisible by 128 (exactly zero for this problem).
    if (blockIdx.x == 0 && waveInBlock == 0) {
        for (long long i = (nChunks << 7) + lane; i < n; i += WAVE)
            d += x[i] * y[i];
    }

    // Wave reduction (wave32).
#pragma unroll
    for (int off = 16; off > 0; off >>= 1)
        d += __shfl_xor(d, off, WAVE);

    __shared__ float smem[8];
    if (lane == 0) smem[waveInBlock] = d;
    __syncthreads();
    if (waveInBlock == 0) {
        float w = (lane < wavesPerBlk) ? smem[lane] : 0.f;
#pragma unroll
        for (int off = 16; off > 0; off >>= 1)
            w += __shfl_xor(w, off, WAVE);
        if (lane == 0) partial[blockIdx.x] = w;   // written unconditionally
    }
}

// Single-block deterministic final reduction; writes -sum to out.
__global__ __launch_bounds__(256) void cc_final_reduce_kernel(
    const float* __restrict__ partial, int nPartials, float* __restrict__ out)
{
    float s = 0.f;
    for (int i = threadIdx.x; i < nPartials; i += blockDim.x)
        s += partial[i];

    const int lane = threadIdx.x & (WAVE - 1);
    const int w    = (int)(threadIdx.x >> 5);
#pragma unroll
    for (int off = 16; off > 0; off >>= 1)
        s += __shfl_xor(s, off, WAVE);

    __shared__ float smem[8];
    if (lane == 0) smem[w] = s;
    __syncthreads();
    if (w == 0) {
        float t = (lane < 8) ? smem[lane] : 0.f;
#pragma unroll
        for (int off = 16; off > 0; off >>= 1)
            t += __shfl_xor(t, off, WAVE);
        if (lane == 0) *out = -t;
    }
}

extern "C" void kernel_launch(void* const* d_in, const int* in_sizes, int n_in,
                              void* d_out, int out_size, void* d_ws, size_t ws_size,
                              hipStream_t stream) {
    const float* x   = (const float*)d_in[0];
    const float* y   = (const float*)d_in[1];
    float*       out = (float*)d_out;
    float*       partial = (float*)d_ws;

    const long long n = (long long)in_sizes[0];
    const long long nChunks = n >> 7;

    long long want = (nChunks + 7) / 8;        // ~8 chunks per block (~1 per wave) before the cap
    if (want < 1) want = 1;
    int blocks = (int)(want < 4096 ? want : 4096);

    cc_dot_wmma_kernel<<<blocks, 256, 0, stream>>>(x, y, partial, n);
    cc_final_reduce_kernel<<<1, 256, 0, stream>>>(partial, blocks, out);
}